// GINEncoder_86509231276725
// MI455X (gfx1250) — compile-verified
//
#include <hip/hip_runtime.h>

#define N_NODES 100000
#define N_EDGES 1280000
#define D 64
#define D2 128
#define BN_EPS 1e-5f

typedef __attribute__((ext_vector_type(2))) float v2f;
typedef __attribute__((ext_vector_type(8))) float v8f;

__device__ __forceinline__ v8f wmma4(v2f a, v2f b, v8f c) {
  // V_WMMA_F32_16X16X4_F32 : D = A(16x4) * B(4x16) + C(16x16), full f32
  return __builtin_amdgcn_wmma_f32_16x16x4_f32(
      /*neg_a=*/false, a, /*neg_b=*/false, b,
      /*c_mod=*/(short)0, c, /*reuse_a=*/false, /*reuse_b=*/false);
}

// ---------------------------------------------------------------- utilities
__global__ __launch_bounds__(256) void zero_kernel(float* __restrict__ p, int n) {
  int tid = blockIdx.x * 256 + threadIdx.x;
  if (tid < n) p[tid] = 0.0f;
}

// ------------------------------------------------------- input projection
// h = x @ lin_w + lin_b   ([N,9] @ [9,64])
__global__ __launch_bounds__(256) void lin_kernel(const float* __restrict__ x,
                                                  const float* __restrict__ w,
                                                  const float* __restrict__ b,
                                                  float* __restrict__ h) {
  int tid = blockIdx.x * 256 + threadIdx.x;
  if (tid >= N_NODES * D) return;
  int n = tid >> 6, d = tid & 63;
  float acc = b[d];
#pragma unroll
  for (int k = 0; k < 9; ++k) acc += x[n * 9 + k] * w[k * D + d];
  h[tid] = acc;
}

// ------------------------------------------------------- edge messages
// msg = relu(h[src] + sum_f emb[f][attr_f]) ; agg[dst] += msg
__global__ __launch_bounds__(256) void message_kernel(const float* __restrict__ h,
                                                      const float* __restrict__ emb, // [3][5][64]
                                                      const int* __restrict__ src,
                                                      const int* __restrict__ dst,
                                                      const int* __restrict__ attr, // [E][3]
                                                      float* __restrict__ agg) {
  int t = threadIdx.x;
  int c = t & 63;
  int e = (blockIdx.x << 2) + (t >> 6);
  if (e >= N_EDGES) return;
  int s  = src[e];
  int d  = dst[e];
  int a0 = attr[e * 3 + 0];
  int a1 = attr[e * 3 + 1];
  int a2 = attr[e * 3 + 2];
  float ev = emb[(0 * 5 + a0) * D + c] + emb[(1 * 5 + a1) * D + c] + emb[(2 * 5 + a2) * D + c];
  float msg = fmaxf(h[s * D + c] + ev, 0.0f);
  unsafeAtomicAdd(&agg[d * D + c], msg);  // native global_atomic_add_f32
}

// ------------------------------------------------------- GEMM1 (WMMA f32)
// t1 = (h + agg) @ W[64,128] + bias ; block = 32 rows x 128 cols, 8 waves
__global__ __launch_bounds__(256) void gemm1_kernel(const float* __restrict__ h,
                                                    const float* __restrict__ agg,
                                                    const float* __restrict__ W,
                                                    const float* __restrict__ bias,
                                                    float* __restrict__ out) {
  __shared__ float Alds[32 * 68];    // stride 68: banks (4m+k) conflict-free for WMMA A reads
  __shared__ float Blds[64 * 136];   // stride 136: banks (8k+n) conflict-free for WMMA B reads
  int t = threadIdx.x;
  int rowBase = blockIdx.x * 32;

  for (int i = t; i < 32 * 64; i += 256) {           // A tile = h + agg (fused)
    int r = i >> 6, c = i & 63;
    int g = (rowBase + r) * D + c;
    Alds[r * 68 + c] = h[g] + agg[g];
  }
  for (int i = t; i < 64 * 128; i += 256) {          // B = W (shared by whole block)
    int r = i >> 7, c = i & 127;
    Blds[r * 136 + c] = W[i];
  }
  __syncthreads();

  int lane = t & 31;
  int wave = t >> 5;          // col tile 0..7 (16 cols each)
  int m    = lane & 15;       // A row within tile / B col within tile
  int half = lane >> 4;       // K-pair select
  int koff = half << 1;

  v8f acc0 = {};
  v8f acc1 = {};
#pragma unroll
  for (int kk = 0; kk < 16; ++kk) {
    int k = (kk << 2) + koff;
    v2f b, a0, a1;
    b[0]  = Blds[k * 136 + wave * 16 + m];
    b[1]  = Blds[(k + 1) * 136 + wave * 16 + m];
    a0[0] = Alds[m * 68 + k];
    a0[1] = Alds[m * 68 + k + 1];
    a1[0] = Alds[(16 + m) * 68 + k];
    a1[1] = Alds[(16 + m) * 68 + k + 1];
    acc0 = wmma4(a0, b, acc0);
    acc1 = wmma4(a1, b, acc1);
  }

  int col  = wave * 16 + m;
  float bv = bias[col];
#pragma unroll
  for (int j = 0; j < 8; ++j) {
    int r = rowBase + j + (half << 3);
    out[r * D2 + col]        = acc0[j] + bv;
    out[(r + 16) * D2 + col] = acc1[j] + bv;
  }
}

// ------------------------------------------------------- GEMM2 (WMMA f32)
// t2 = relu(BN(t1)) @ W[128,64] + bias ; BN apply fused into A-tile load
__global__ __launch_bounds__(256) void gemm2_kernel(const float* __restrict__ t1,
                                                    const float* __restrict__ W,
                                                    const float* __restrict__ bias,
                                                    const float* __restrict__ gamma,
                                                    const float* __restrict__ beta,
                                                    const float* __restrict__ sum,
                                                    const float* __restrict__ sq,
                                                    float* __restrict__ out) {
  __shared__ float Alds[32 * 132];   // stride 132 (≡4 mod 64)
  __shared__ float Blds[128 * 72];   // stride 72  (≡8 mod 64)
  __shared__ float scale[D2], shift[D2];
  int t = threadIdx.x;
  if (t < D2) {
    float mu  = sum[t] * (1.0f / N_NODES);
    float var = sq[t] * (1.0f / N_NODES) - mu * mu;
    float sc  = gamma[t] * rsqrtf(var + BN_EPS);
    scale[t] = sc;
    shift[t] = beta[t] - mu * sc;
  }
  __syncthreads();

  int rowBase = blockIdx.x * 32;
  for (int i = t; i < 32 * 128; i += 256) {          // A = relu(BN(t1)) (fused)
    int r = i >> 7, c = i & 127;
    float v = t1[(rowBase + r) * D2 + c] * scale[c] + shift[c];
    Alds[r * 132 + c] = fmaxf(v, 0.0f);
  }
  for (int i = t; i < 128 * 64; i += 256) {          // B = W
    int r = i >> 6, c = i & 63;
    Blds[r * 72 + c] = W[i];
  }
  __syncthreads();

  int lane = t & 31;
  int wave = t >> 5;
  int ct   = wave & 3;        // col tile 0..3
  int rt   = wave >> 2;       // row tile 0..1
  int m    = lane & 15;
  int half = lane >> 4;
  int koff = half << 1;

  v8f acc = {};
#pragma unroll
  for (int kk = 0; kk < 32; ++kk) {
    int k = (kk << 2) + koff;
    v2f a, b;
    a[0] = Alds[(rt * 16 + m) * 132 + k];
    a[1] = Alds[(rt * 16 + m) * 132 + k + 1];
    b[0] = Blds[k * 72 + ct * 16 + m];
    b[1] = Blds[(k + 1) * 72 + ct * 16 + m];
    acc = wmma4(a, b, acc);
  }

  int col  = ct * 16 + m;
  float bv = bias[col];
#pragma unroll
  for (int j = 0; j < 8; ++j) {
    int r = rowBase + rt * 16 + j + (half << 3);
    out[r * D + col] = acc[j] + bv;
  }
}

// ------------------------------------------------------- column stats (sum, sumsq)
template <int C>
__global__ __launch_bounds__(256) void stats_kernel(const float* __restrict__ x,
                                                    float* __restrict__ sum,
                                                    float* __restrict__ sq) {
  constexpr int RPB = 256 / C;
  int t  = threadIdx.x;
  int c  = t % C;
  int rg = t / C;
  float s = 0.0f, q = 0.0f;
  for (int n = blockIdx.x * RPB + rg; n < N_NODES; n += gridDim.x * RPB) {
    float v = x[n * C + c];
    s += v;
    q += v * v;
  }
  __shared__ float ls[256], lq[256];
  ls[t] = s;
  lq[t] = q;
  __syncthreads();
  if (t < C) {
#pragma unroll
    for (int j = 1; j < RPB; ++j) {
      s += ls[t + j * C];
      q += lq[t + j * C];
    }
    unsafeAtomicAdd(&sum[c], s);
    unsafeAtomicAdd(&sq[c], q);
  }
}

// ------------------------------------------------------- final BN (+relu)
__global__ __launch_bounds__(256) void bn_apply_kernel(const float* __restrict__ t2,
                                                       const float* __restrict__ gamma,
                                                       const float* __restrict__ beta,
                                                       const float* __restrict__ sum,
                                                       const float* __restrict__ sq,
                                                       int do_relu,
                                                       float* __restrict__ h) {
  int tid = blockIdx.x * 256 + threadIdx.x;
  if (tid >= N_NODES * D) return;
  int c = tid & 63;
  float mu  = sum[c] * (1.0f / N_NODES);
  float var = sq[c] * (1.0f / N_NODES) - mu * mu;
  float sc  = gamma[c] * rsqrtf(var + BN_EPS);
  float v   = t2[tid] * sc + (beta[c] - mu * sc);
  h[tid] = do_relu ? fmaxf(v, 0.0f) : v;
}

extern "C" void kernel_launch(void* const* d_in, const int* in_sizes, int n_in,
                              void* d_out, int out_size, void* d_ws, size_t ws_size,
                              hipStream_t stream) {
  const float* x     = (const float*)d_in[0];
  const float* lin_w = (const float*)d_in[1];
  const float* lin_b = (const float*)d_in[2];
  const float* bond  = (const float*)d_in[3];   // [3][3][5][64]
  const float* l1w   = (const float*)d_in[4];   // [3][64][128]
  const float* l1b   = (const float*)d_in[5];   // [3][128]
  const float* mg    = (const float*)d_in[6];   // [3][128]
  const float* mb    = (const float*)d_in[7];   // [3][128]
  const float* l2w   = (const float*)d_in[8];   // [3][128][64]
  const float* l2b   = (const float*)d_in[9];   // [3][64]
  const float* bg    = (const float*)d_in[10];  // [3][64]
  const float* bbp   = (const float*)d_in[11];  // [3][64]
  const int*   ei    = (const int*)d_in[12];    // [2][E]
  const int*   ea    = (const int*)d_in[13];    // [E][3]

  float* h = (float*)d_out;                       // [N,64] doubles as the live h buffer

  float* agg  = (float*)d_ws;                     // [N,64]
  float* t1   = agg + (size_t)N_NODES * D;        // [N,128]
  float* stat = t1 + (size_t)N_NODES * D2;        // 384 floats: sum1[128] sq1[128] sum2[64] sq2[64]
  float* t2   = agg;                              // alias: agg dead after gemm1

  const int elemN = N_NODES * D;

  lin_kernel<<<(elemN + 255) / 256, 256, 0, stream>>>(x, lin_w, lin_b, h);

  for (int i = 0; i < 3; ++i) {
    zero_kernel<<<(elemN + 255) / 256, 256, 0, stream>>>(agg, elemN);
    zero_kernel<<<2, 256, 0, stream>>>(stat, 384);

    message_kernel<<<N_EDGES / 4, 256, 0, stream>>>(
        h, bond + (size_t)i * 3 * 5 * D, ei, ei + N_EDGES, ea, agg);

    gemm1_kernel<<<N_NODES / 32, 256, 0, stream>>>(
        h, agg, l1w + (size_t)i * D * D2, l1b + (size_t)i * D2, t1);

    stats_kernel<128><<<512, 256, 0, stream>>>(t1, stat, stat + 128);

    gemm2_kernel<<<N_NODES / 32, 256, 0, stream>>>(
        t1, l2w + (size_t)i * D2 * D, l2b + (size_t)i * D,
        mg + (size_t)i * D2, mb + (size_t)i * D2, stat, stat + 128, t2);

    stats_kernel<64><<<512, 256, 0, stream>>>(t2, stat + 256, stat + 320);

    bn_apply_kernel<<<(elemN + 255) / 256, 256, 0, stream>>>(
        t2, bg + (size_t)i * D, bbp + (size_t)i * D, stat + 256, stat + 320,
        (i < 2) ? 1 : 0, h);
  }
}